// MTSiLU_68530498175130
// MI455X (gfx1250) — compile-verified
//
#include <hip/hip_runtime.h>

// MTLU + SiLU fused elementwise kernel for MI455X (gfx1250, wave32).
// x: [B=16, C=64, H=256, W=256] f32; weight/bias: [C=64, BIN=20] f32.
// out[i] = (w[c,bin]*x + b[c,bin]) * sigmoid(x),
// bin = clamp(floor(x*10)+10, 0, 19), c = (i >> 16) & 63.
//
// Pure HBM-bandwidth kernel (512 MB traffic, ~0.7 GFLOP): b128 vector
// loads/stores, per-block packed {w,b} table in LDS (one ds_load_b64 per
// element), speculative global_prefetch, async global->LDS table staging.

#define BIN_NUM   20
#define HW_SHIFT  16                      // H*W = 65536 = 1 << 16
#define BLOCK     256                     // 8 wave32 waves
#define VEC       4                       // float4 / b128
#define ITERS     16
#define CHUNK     (BLOCK * VEC * ITERS)   // 16384 elems/block, divides H*W

#if __has_builtin(__builtin_amdgcn_global_load_async_to_lds_b32)
#define HAVE_ASYNC_LDS 1
#else
#define HAVE_ASYNC_LDS 0
#endif

typedef __attribute__((address_space(1))) int g_int;   // global
typedef __attribute__((address_space(3))) int l_int;   // LDS

__device__ __forceinline__ float mtsilu_elem(float v, const float2* tab) {
    int bin = (int)floorf(v * 10.0f) + (BIN_NUM / 2);
    bin = min(max(bin, 0), BIN_NUM - 1);
    const float2 wb = tab[bin];                                 // ds_load_b64
    const float sig = __builtin_amdgcn_rcpf(1.0f + __expf(-v)); // v_exp+v_rcp
    return fmaf(wb.x, v, wb.y) * sig;
}

__global__ __launch_bounds__(BLOCK) void mtsilu_kernel(
    const float* __restrict__ x,
    const float* __restrict__ w,
    const float* __restrict__ b,
    float* __restrict__ out,
    int n_elems,
    int ch_mask)     // n_channels - 1 (power of two)
{
    __shared__ float2 tab[BIN_NUM];

    const long long base = (long long)blockIdx.x * CHUNK;
    // CHUNK divides H*W, so the channel is uniform across this block.
    const int ch  = ((int)(base >> HW_SHIFT)) & ch_mask;
    const int tid = threadIdx.x;

#if HAVE_ASYNC_LDS
    // gfx1250 async DMA path: global -> LDS without touching VGPRs.
    __shared__ float staging[2 * BIN_NUM];
    if (tid < BIN_NUM) {
        __builtin_amdgcn_global_load_async_to_lds_b32(
            (g_int*)(w + ch * BIN_NUM + tid),
            (l_int*)&staging[tid], 0, 0);
        __builtin_amdgcn_global_load_async_to_lds_b32(
            (g_int*)(b + ch * BIN_NUM + tid),
            (l_int*)&staging[BIN_NUM + tid], 0, 0);
    }
#if __has_builtin(__builtin_amdgcn_s_wait_asynccnt)
    __builtin_amdgcn_s_wait_asynccnt(0);
#else
    asm volatile("s_wait_asynccnt 0x0" ::: "memory");
#endif
    __syncthreads();
    if (tid < BIN_NUM) {
        tab[tid] = make_float2(staging[tid], staging[BIN_NUM + tid]);
    }
#else
    if (tid < BIN_NUM) {
        tab[tid] = make_float2(w[ch * BIN_NUM + tid], b[ch * BIN_NUM + tid]);
    }
#endif
    __syncthreads();

    const float* xp = x + base;      // one scalar 64-bit add; inner offsets 32-bit
    float*       op = out + base;

    if (base + CHUNK <= (long long)n_elems) {
        // Fast path: whole chunk in range (always taken for 16x64x256x256).
#pragma unroll
        for (int it = 0; it < ITERS; ++it) {
            const int idx = (it * BLOCK + tid) * VEC;
            __builtin_prefetch(xp + idx + 4 * BLOCK * VEC, 0, 1);  // global_prefetch
            const float4 xv = *(const float4*)(xp + idx);          // global_load_b128
            float4 yv;
            yv.x = mtsilu_elem(xv.x, tab);
            yv.y = mtsilu_elem(xv.y, tab);
            yv.z = mtsilu_elem(xv.z, tab);
            yv.w = mtsilu_elem(xv.w, tab);
            *(float4*)(op + idx) = yv;                             // global_store_b128
        }
    } else {
        // Guarded tail path (unused for the reference shape).
        for (int it = 0; it < ITERS; ++it) {
            const int idx = (it * BLOCK + tid) * VEC;
            if (base + idx + (VEC - 1) < (long long)n_elems) {
                const float4 xv = *(const float4*)(xp + idx);
                float4 yv;
                yv.x = mtsilu_elem(xv.x, tab);
                yv.y = mtsilu_elem(xv.y, tab);
                yv.z = mtsilu_elem(xv.z, tab);
                yv.w = mtsilu_elem(xv.w, tab);
                *(float4*)(op + idx) = yv;
            }
        }
    }
}

extern "C" void kernel_launch(void* const* d_in, const int* in_sizes, int n_in,
                              void* d_out, int out_size, void* d_ws, size_t ws_size,
                              hipStream_t stream) {
    const float* x  = (const float*)d_in[0];
    const float* w  = (const float*)d_in[1];
    const float* bb = (const float*)d_in[2];
    float* out = (float*)d_out;

    const int n          = in_sizes[0];                 // 67,108,864
    const int n_channels = in_sizes[1] / BIN_NUM;       // 64 (power of two)
    const int ch_mask    = n_channels - 1;

    const int grid = (n + CHUNK - 1) / CHUNK;           // 4096 blocks
    mtsilu_kernel<<<grid, BLOCK, 0, stream>>>(x, w, bb, out, n, ch_mask);
}